// Recommender_9964324127510
// MI455X (gfx1250) — compile-verified
//
#include <hip/hip_runtime.h>

#define N_ENTITIES 100000
#define N_USERS    50000
#define CHANNEL    128
#define N_EDGESC   2000000
#define NNZC       1000000
#define R0C        42033
#define R1C        44630
#define RWC        (R1C - R0C)   // 2597
#define N_HOPS     3

typedef __attribute__((ext_vector_type(2))) float v2f;
typedef __attribute__((ext_vector_type(8))) float v8f;

// ---------------------------------------------------------------------------
// init: ent working copy = entity_emb; out_ent = entity_emb; out_usr = user_emb
// ---------------------------------------------------------------------------
__global__ __launch_bounds__(256)
void kgat_init(const float4* __restrict__ ent_in, const float4* __restrict__ usr_in,
               float4* __restrict__ ent, float4* __restrict__ out_ent,
               float4* __restrict__ out_usr) {
    int i = blockIdx.x * blockDim.x + threadIdx.x;
    const int NE4 = N_ENTITIES * CHANNEL / 4;
    const int NU4 = N_USERS * CHANNEL / 4;
    if (i < NE4) { float4 v = ent_in[i]; ent[i] = v; out_ent[i] = v; }
    if (i < NU4) { out_usr[i] = usr_in[i]; }
}

__global__ __launch_bounds__(256)
void kgat_zero(float4* __restrict__ p, int n4) {
    int i = blockIdx.x * blockDim.x + threadIdx.x;
    if (i < n4) p[i] = make_float4(0.f, 0.f, 0.f, 0.f);
}

// ---------------------------------------------------------------------------
// Region GEMM via V_WMMA_F32_16X16X4_F32 (fp32 in, fp32 acc).
// region[M=2597, N=128] = rw[2597 x 2597] * ent[R0: , 0:128]
// Block = 256 threads = 8 waves; wave w handles N-tile w (N=128 = 8*16).
// blockIdx.x = M tile (163 tiles of 16).
//
// Hot loop is guard-free: K remainder (2597 % 4 == 1) handled in a branchless
// epilogue (clamped address + value select), M edge handled by clamping the
// A-row pointer (out-of-range rows are computed but never stored).
// ---------------------------------------------------------------------------
__global__ __launch_bounds__(256)
void kgat_region_gemm(const float* __restrict__ rw,
                      const float* __restrict__ ent,
                      float* __restrict__ region) {
    const int lane = threadIdx.x & 31;
    const int nt   = threadIdx.x >> 5;     // 0..7  (N tile)
    const int mt   = blockIdx.x;           // 0..162 (M tile)
    const int mn   = lane & 15;            // m within A frag == n within B/C frag
    const int kh   = lane >> 4;            // lane half

    const int m    = mt * 16 + mn;
    const int mC   = (m < RWC) ? m : (RWC - 1);     // clamp: dup row, never stored
    const int ncol = nt * 16 + mn;
    const float* __restrict__ arow = rw + (size_t)mC * RWC;
    const float* __restrict__ bcol = ent + (size_t)R0C * CHANNEL + ncol;

    v8f acc = {};
    const int KMAIN = RWC & ~3;            // 2596; max index ka+1 = 2595 < RWC
    for (int k = 0; k < KMAIN; k += 4) {
        const int ka = k + 2 * kh;
        v2f a, b;
        a.x = arow[ka];
        a.y = arow[ka + 1];
        b.x = bcol[(size_t)ka * CHANNEL];
        b.y = bcol[(size_t)(ka + 1) * CHANNEL];
        // 8 args: (neg_a, A, neg_b, B, c_mod, C, reuse_a, reuse_b)
        acc = __builtin_amdgcn_wmma_f32_16x16x4_f32(
            false, a, false, b, (short)0, acc, false, false);
    }
    {   // K remainder (k = 2596): unconditional clamped loads + cndmask to zero
        const int ka  = KMAIN + 2 * kh;
        const int ka0 = (ka     < RWC) ? ka       : (RWC - 1);
        const int ka1 = (ka + 1 < RWC) ? (ka + 1) : (RWC - 1);
        const float ax = arow[ka0];
        const float ay = arow[ka1];
        const float bx = bcol[(size_t)ka0 * CHANNEL];
        const float by = bcol[(size_t)ka1 * CHANNEL];
        v2f a, b;
        a.x = (ka     < RWC) ? ax : 0.f;
        a.y = (ka + 1 < RWC) ? ay : 0.f;
        b.x = (ka     < RWC) ? bx : 0.f;
        b.y = (ka + 1 < RWC) ? by : 0.f;
        acc = __builtin_amdgcn_wmma_f32_16x16x4_f32(
            false, a, false, b, (short)0, acc, false, false);
    }
    // C/D layout: VGPR v, lane half kh -> row (v + 8*kh), col = lane&15
    #pragma unroll
    for (int v = 0; v < 8; ++v) {
        int row = mt * 16 + v + 8 * kh;
        if (row < RWC) region[(size_t)row * CHANNEL + ncol] = acc[v];
    }
}

// ent[R0+r][c] = 0.8*ent + 0.2*region  (rows outside region are unchanged:
// 0.8x + 0.2x == x in the reference)
__global__ __launch_bounds__(256)
void kgat_blend(float* __restrict__ ent, const float* __restrict__ region) {
    int i = blockIdx.x * blockDim.x + threadIdx.x;
    if (i < RWC * CHANNEL) {
        int r = i / CHANNEL, c = i - r * CHANNEL;
        size_t idx = (size_t)(R0C + r) * CHANNEL + c;
        ent[idx] = 0.8f * ent[idx] + 0.2f * region[i];
    }
}

// ---------------------------------------------------------------------------
// KG aggregate: one wave per edge; lane holds float4 (4 channels).
// agg[head] += ent[tail] * weight[type-1]   (inv_deg cancels under l2-norm)
// ---------------------------------------------------------------------------
__global__ __launch_bounds__(256)
void kgat_kg_agg(const int* __restrict__ head, const int* __restrict__ tail,
                 const int* __restrict__ etype, const float* __restrict__ wrel,
                 const float* __restrict__ ent, float* __restrict__ agg) {
    const int lane = threadIdx.x & 31;
    const int e = blockIdx.x * (blockDim.x >> 5) + (threadIdx.x >> 5);
    if (e >= N_EDGESC) return;
    const int h = head[e];
    const int t = tail[e];
    const int ty = etype[e] - 1;
    float4 v = ((const float4*)(ent  + (size_t)t  * CHANNEL))[lane];
    float4 w = ((const float4*)(wrel + (size_t)ty * CHANNEL))[lane];
    float* dst = agg + (size_t)h * CHANNEL + lane * 4;
    atomicAdd(dst + 0, v.x * w.x);
    atomicAdd(dst + 1, v.y * w.y);
    atomicAdd(dst + 2, v.z * w.z);
    atomicAdd(dst + 3, v.w * w.w);
}

// user aggregate: uagg[row] += val * ent[col]
__global__ __launch_bounds__(256)
void kgat_user_agg(const int* __restrict__ rows, const int* __restrict__ cols,
                   const float* __restrict__ vals, const float* __restrict__ ent,
                   float* __restrict__ uagg) {
    const int lane = threadIdx.x & 31;
    const int e = blockIdx.x * (blockDim.x >> 5) + (threadIdx.x >> 5);
    if (e >= NNZC) return;
    const int r = rows[e];
    const int c = cols[e];
    const float s = vals[e];
    float4 v = ((const float4*)(ent + (size_t)c * CHANNEL))[lane];
    float* dst = uagg + (size_t)r * CHANNEL + lane * 4;
    atomicAdd(dst + 0, s * v.x);
    atomicAdd(dst + 1, s * v.y);
    atomicAdd(dst + 2, s * v.z);
    atomicAdd(dst + 3, s * v.w);
}

// ---------------------------------------------------------------------------
// normalize + residual: wave per row, lane holds float4, wave32 xor-reduction
// res[row] += normalize(agg[row]); if (next) next[row] = normalize(agg[row])
// ---------------------------------------------------------------------------
__global__ __launch_bounds__(256)
void kgat_norm_res(const float* __restrict__ agg, float* __restrict__ next,
                   float* __restrict__ res, int nrows) {
    const int lane = threadIdx.x & 31;
    const int row = blockIdx.x * (blockDim.x >> 5) + (threadIdx.x >> 5);
    if (row >= nrows) return;
    float4 v = ((const float4*)(agg + (size_t)row * CHANNEL))[lane];
    float ss = v.x * v.x + v.y * v.y + v.z * v.z + v.w * v.w;
    #pragma unroll
    for (int off = 16; off > 0; off >>= 1) ss += __shfl_xor(ss, off, 32);
    const float inv = 1.0f / fmaxf(sqrtf(ss), 1e-12f);
    v.x *= inv; v.y *= inv; v.z *= inv; v.w *= inv;
    if (next) ((float4*)(next + (size_t)row * CHANNEL))[lane] = v;
    float4* rp = (float4*)(res + (size_t)row * CHANNEL);
    float4 cur = rp[lane];
    cur.x += v.x; cur.y += v.y; cur.z += v.z; cur.w += v.w;
    rp[lane] = cur;
}

// ---------------------------------------------------------------------------
extern "C" void kernel_launch(void* const* d_in, const int* in_sizes, int n_in,
                              void* d_out, int out_size, void* d_ws, size_t ws_size,
                              hipStream_t stream) {
    (void)in_sizes; (void)n_in; (void)out_size; (void)ws_size;
    const float* user_emb   = (const float*)d_in[0];
    const float* entity_emb = (const float*)d_in[1];
    const float* rwm        = (const float*)d_in[2];
    const float* weight     = (const float*)d_in[3];
    const float* ivals      = (const float*)d_in[4];
    const int*   ehead      = (const int*)d_in[5];
    const int*   etail      = (const int*)d_in[6];
    const int*   etype      = (const int*)d_in[7];
    const int*   irows      = (const int*)d_in[8];
    const int*   icols      = (const int*)d_in[9];

    float* out_ent = (float*)d_out;
    float* out_usr = out_ent + (size_t)N_ENTITIES * CHANNEL;

    float* ws     = (float*)d_ws;
    float* ent    = ws;  ws += (size_t)N_ENTITIES * CHANNEL;   // 12.8M
    float* agg    = ws;  ws += (size_t)N_ENTITIES * CHANNEL;   // 12.8M
    float* uagg   = ws;  ws += (size_t)N_USERS * CHANNEL;      // 6.4M (contiguous after agg)
    float* region = ws;                                        // 2597*128

    const int NE4 = N_ENTITIES * CHANNEL / 4;                  // 3.2M
    const int AGG4 = (N_ENTITIES + N_USERS) * CHANNEL / 4;     // agg+uagg together

    kgat_init<<<(NE4 + 255) / 256, 256, 0, stream>>>(
        (const float4*)entity_emb, (const float4*)user_emb,
        (float4*)ent, (float4*)out_ent, (float4*)out_usr);

    const int mtiles = (RWC + 15) / 16;                        // 163
    for (int hop = 0; hop < N_HOPS; ++hop) {
        kgat_zero<<<(AGG4 + 255) / 256, 256, 0, stream>>>((float4*)agg, AGG4);
        kgat_region_gemm<<<mtiles, 256, 0, stream>>>(rwm, ent, region);
        kgat_blend<<<(RWC * CHANNEL + 255) / 256, 256, 0, stream>>>(ent, region);
        kgat_kg_agg<<<N_EDGESC / 8, 256, 0, stream>>>(ehead, etail, etype, weight, ent, agg);
        kgat_user_agg<<<NNZC / 8, 256, 0, stream>>>(irows, icols, ivals, ent, uagg);
        kgat_norm_res<<<N_ENTITIES / 8, 256, 0, stream>>>(agg, ent, out_ent, N_ENTITIES);
        kgat_norm_res<<<N_USERS / 8, 256, 0, stream>>>(uagg, nullptr, out_usr, N_USERS);
    }
}